// RelationalReasoningModuleWithLoop_33741263078016
// MI455X (gfx1250) — compile-verified
//
#include <hip/hip_runtime.h>

typedef float v2f __attribute__((ext_vector_type(2)));
typedef float v8f __attribute__((ext_vector_type(8)));

#define BATCH 32
#define DCH   64
#define PX    144      // 12*12
#define ECH   128
#define HID   256
#define OUTD  256
#define NPAIRS 10296   // 144*143/2
#define HC    32       // h-columns per pair-sum block
#define NG    32       // j-groups per pair-sum block

// ---------------------------------------------------------------------------
// Kernel 1: A[b,p,h]  = xf[b,p,:] @ W1[0:64, h]   (xf[p][d] = x[b][d][p])
//           Bm[b,p,h] = xf[b,p,:] @ W1[64:128, h]
// One wave per 16x16 (p,h) tile; A-fragment of xf shared by both products.
// ---------------------------------------------------------------------------
__global__ __launch_bounds__(32) void k_layer1_x(
    const float* __restrict__ x,    // [B, D, PX]
    const float* __restrict__ W1,   // [384, HID]
    float* __restrict__ Aout,       // [B, PX, HID]
    float* __restrict__ Bout)       // [B, PX, HID]
{
    const int lane = threadIdx.x;
    const int n0 = blockIdx.x * 16;         // HID tile
    const int p0 = blockIdx.y * 16;         // PX tile
    const int b  = blockIdx.z;
    const int MN = lane & 15;               // row (A) / col (B) within tile
    const int kh = (lane >> 4) << 1;        // K sub-offset: 0 or 2

    const float* xb = x + (size_t)b * DCH * PX;
    v8f accA = {0.f,0.f,0.f,0.f,0.f,0.f,0.f,0.f};
    v8f accB = {0.f,0.f,0.f,0.f,0.f,0.f,0.f,0.f};

    for (int kb = 0; kb < DCH; kb += 4) {
        const int k0 = kb + kh;
        v2f a;
        a.x = xb[(size_t)(k0 + 0) * PX + p0 + MN];
        a.y = xb[(size_t)(k0 + 1) * PX + p0 + MN];
        v2f w0, w1;
        w0.x = W1[(size_t)(k0 + 0) * HID + n0 + MN];
        w0.y = W1[(size_t)(k0 + 1) * HID + n0 + MN];
        w1.x = W1[(size_t)(DCH + k0 + 0) * HID + n0 + MN];
        w1.y = W1[(size_t)(DCH + k0 + 1) * HID + n0 + MN];
        accA = __builtin_amdgcn_wmma_f32_16x16x4_f32(false, a, false, w0,
                                                     (short)0, accA, false, false);
        accB = __builtin_amdgcn_wmma_f32_16x16x4_f32(false, a, false, w1,
                                                     (short)0, accB, false, false);
    }

    const int hi = lane >> 4;               // 0 or 1 -> M = r or 8+r
    const size_t base = (size_t)b * PX * HID;
#pragma unroll
    for (int r = 0; r < 8; ++r) {
        const int row = p0 + r + 8 * hi;
        Aout[base + (size_t)row * HID + n0 + MN] = accA[r];
        Bout[base + (size_t)row * HID + n0 + MN] = accB[r];
    }
}

// ---------------------------------------------------------------------------
// Kernel 2: Cemb[b,h] = emb[b,:] @ W1[128:256, h] + b1[h]
// ---------------------------------------------------------------------------
__global__ __launch_bounds__(32) void k_layer1_emb(
    const float* __restrict__ emb,  // [B, E]
    const float* __restrict__ W1,   // [384, HID]
    const float* __restrict__ b1,   // [HID]
    float* __restrict__ Cemb)       // [B, HID]
{
    const int lane = threadIdx.x;
    const int n0 = blockIdx.x * 16;
    const int m0 = blockIdx.y * 16;         // batch tile
    const int MN = lane & 15;
    const int kh = (lane >> 4) << 1;

    v8f acc = {0.f,0.f,0.f,0.f,0.f,0.f,0.f,0.f};
    for (int kb = 0; kb < ECH; kb += 4) {
        const int k0 = kb + kh;
        v2f a;
        a.x = emb[(size_t)(m0 + MN) * ECH + k0 + 0];
        a.y = emb[(size_t)(m0 + MN) * ECH + k0 + 1];
        v2f w;
        w.x = W1[(size_t)(2 * DCH + k0 + 0) * HID + n0 + MN];
        w.y = W1[(size_t)(2 * DCH + k0 + 1) * HID + n0 + MN];
        acc = __builtin_amdgcn_wmma_f32_16x16x4_f32(false, a, false, w,
                                                    (short)0, acc, false, false);
    }

    const int hi = lane >> 4;
    const float bias = b1[n0 + MN];
#pragma unroll
    for (int r = 0; r < 8; ++r) {
        const int row = m0 + r + 8 * hi;
        Cemb[(size_t)row * HID + n0 + MN] = acc[r] + bias;
    }
}

// ---------------------------------------------------------------------------
// Kernel 3: S[b,h] = sum_{i<j} relu(A[b,i,h] + Bm[b,j,h] + Cemb[b,h])
// Block: 256 threads = 32 j-groups x 8 column-quads (float4 -> 32 h-columns).
// A/Bm slabs staged in LDS; inner loop reads ds_load_b128 per 4 elements.
// ---------------------------------------------------------------------------
__global__ __launch_bounds__(256) void k_pairsum(
    const float* __restrict__ Ag,   // [B, PX, HID]
    const float* __restrict__ Bg,   // [B, PX, HID]
    const float* __restrict__ Cemb, // [B, HID]
    float* __restrict__ S)          // [B, HID]
{
    __shared__ float4 sA4[PX * (HC / 4)];   // [i][tc] 18 KB
    __shared__ float4 sB4[PX * (HC / 4)];   // 18 KB
    __shared__ float  sPart[NG * HC];       // 4 KB

    const int t  = threadIdx.x;
    const int h0 = blockIdx.x * HC;
    const int b  = blockIdx.y;
    const size_t base = (size_t)b * PX * HID + h0;

    // Stage A/B slabs: 144 rows x 32 cols as float4 (16B-aligned: h0 % 32 == 0)
    for (int idx = t; idx < PX * (HC / 4); idx += 256) {
        const int i  = idx >> 3;            // row 0..143
        const int tc = idx & 7;             // column quad 0..7
        sA4[idx] = ((const float4*)(Ag + base + (size_t)i * HID))[tc];
        sB4[idx] = ((const float4*)(Bg + base + (size_t)i * HID))[tc];
    }
    __syncthreads();

    const int tc  = t & 7;                  // column quad
    const int g   = t >> 3;                 // j-group 0..31
    const float4 cv = *((const float4*)(Cemb + (size_t)b * HID + h0 + tc * 4));

    float4 acc; acc.x = acc.y = acc.z = acc.w = 0.f;
    for (int j = 1 + g; j < PX; j += NG) {
        float4 bj = sB4[j * (HC / 4) + tc];
        bj.x += cv.x; bj.y += cv.y; bj.z += cv.z; bj.w += cv.w;
        for (int i = 0; i < j; ++i) {
            const float4 av = sA4[i * (HC / 4) + tc];
            acc.x += fmaxf(av.x + bj.x, 0.f);
            acc.y += fmaxf(av.y + bj.y, 0.f);
            acc.z += fmaxf(av.z + bj.z, 0.f);
            acc.w += fmaxf(av.w + bj.w, 0.f);
        }
    }
    *((float4*)(sPart + g * HC + tc * 4)) = acc;
    __syncthreads();

    // Reduce 32 partials per column; 32 threads each own one column.
    if (t < HC) {
        float tot = 0.f;
#pragma unroll
        for (int gg = 0; gg < NG; ++gg) tot += sPart[gg * HC + t];
        S[(size_t)b * HID + h0 + t] = tot;
    }
}

// ---------------------------------------------------------------------------
// Kernel 4: out[b,o] = S[b,:] @ W2[:,o] + NPAIRS * b2[o]
// ---------------------------------------------------------------------------
__global__ __launch_bounds__(32) void k_layer2(
    const float* __restrict__ S,    // [B, HID]
    const float* __restrict__ W2,   // [HID, OUT]
    const float* __restrict__ b2,   // [OUT]
    float* __restrict__ out)        // [B, OUT]
{
    const int lane = threadIdx.x;
    const int n0 = blockIdx.x * 16;
    const int m0 = blockIdx.y * 16;
    const int MN = lane & 15;
    const int kh = (lane >> 4) << 1;

    v8f acc = {0.f,0.f,0.f,0.f,0.f,0.f,0.f,0.f};
    for (int kb = 0; kb < HID; kb += 4) {
        const int k0 = kb + kh;
        v2f a;
        a.x = S[(size_t)(m0 + MN) * HID + k0 + 0];
        a.y = S[(size_t)(m0 + MN) * HID + k0 + 1];
        v2f w;
        w.x = W2[(size_t)(k0 + 0) * OUTD + n0 + MN];
        w.y = W2[(size_t)(k0 + 1) * OUTD + n0 + MN];
        acc = __builtin_amdgcn_wmma_f32_16x16x4_f32(false, a, false, w,
                                                    (short)0, acc, false, false);
    }

    const int hi = lane >> 4;
    const float bias = b2[n0 + MN] * (float)NPAIRS;
#pragma unroll
    for (int r = 0; r < 8; ++r) {
        const int row = m0 + r + 8 * hi;
        out[(size_t)row * OUTD + n0 + MN] = acc[r] + bias;
    }
}

// ---------------------------------------------------------------------------
extern "C" void kernel_launch(void* const* d_in, const int* in_sizes, int n_in,
                              void* d_out, int out_size, void* d_ws, size_t ws_size,
                              hipStream_t stream) {
    (void)in_sizes; (void)n_in; (void)out_size; (void)ws_size;

    const float* x   = (const float*)d_in[0];  // [32, 64, 12, 12]
    const float* emb = (const float*)d_in[1];  // [32, 128]
    const float* W1  = (const float*)d_in[2];  // [384, 256]
    const float* b1  = (const float*)d_in[3];  // [256]
    const float* W2  = (const float*)d_in[4];  // [256, 256]
    const float* b2  = (const float*)d_in[5];  // [256]
    float* out = (float*)d_out;                // [32, 256]

    float* Ag   = (float*)d_ws;                       // [B, PX, HID]
    float* Bg   = Ag + (size_t)BATCH * PX * HID;      // [B, PX, HID]
    float* Cemb = Bg + (size_t)BATCH * PX * HID;      // [B, HID]
    float* Ssum = Cemb + (size_t)BATCH * HID;         // [B, HID]

    k_layer1_x  <<<dim3(HID / 16, PX / 16, BATCH), 32, 0, stream>>>(x, W1, Ag, Bg);
    k_layer1_emb<<<dim3(HID / 16, BATCH / 16),     32, 0, stream>>>(emb, W1, b1, Cemb);
    k_pairsum   <<<dim3(HID / HC, BATCH),         256, 0, stream>>>(Ag, Bg, Cemb, Ssum);
    k_layer2    <<<dim3(OUTD / 16, BATCH / 16),    32, 0, stream>>>(Ssum, W2, b2, out);
}